// Conv4d_8057358647869
// MI455X (gfx1250) — compile-verified
//
#include <hip/hip_runtime.h>

// ---------------------------------------------------------------------------
// Conv4d (3,3,3,3) stride1 pad0 as implicit GEMM on CDNA5 WMMA (gfx1250).
//   M = output wo positions (tiles of 16), N = C_OUT (tiles of 16),
//   K = (lk,dk,hk,wk) taps (81) x C_IN (24) = 1944, chunks of 4 via
//   V_WMMA_F32_16X16X4_F32 (exact fp32 -> matches the f32 reference).
// v2: operand pairs are made memory-contiguous so WMMA fragments load
//     directly into even-aligned VGPR pairs (no v_mov shuffling):
//       A: LDS laid out [t3][w][cin-pair], pair = one ds_load_b64
//       B: weights packed row-pair interleaved, pair = one global b64
// ---------------------------------------------------------------------------

typedef __attribute__((ext_vector_type(2))) float v2f;
typedef __attribute__((ext_vector_type(8))) float v8f;

#if defined(__HIP_DEVICE_COMPILE__) && !__has_builtin(__builtin_amdgcn_wmma_f32_16x16x4_f32)
#error "__builtin_amdgcn_wmma_f32_16x16x4_f32 not available on this toolchain"
#endif

#define NBATCH 2
#define CI 24
#define CO 48
#define LOUT 14
#define DOUT 14
#define HOUT 46
#define WOUT 46
#define KTOT (CI * 81)          // 1944

// x strides (floats), layout (B, C_IN, L, D, H, W)
#define HS 48
#define DS2 (48 * 48)           // 2304
#define LS (16 * 48 * 48)       // 36864
#define CS (16 * 16 * 48 * 48)  // 589824
#define BS (24 * 16 * 16 * 48 * 48)

// LDS layout: [t3 (27)][w (50)][cl slots (PADC=10, 8 used)]
#define WP 50
#define PADC 10
#define LDSF (27 * WP * PADC)   // 13500 floats = 54,000 B

// ---------------------------------------------------------------------------
// Pack weight (CO,CI,3,3,3,3) -> bp[(k/2)*CO*2 + n*2 + (k&1)],
// k = tap*24 + cin, tap = lk*27+dk*9+hk*3+wk.  A K-row pair for one n is a
// contiguous, 8B-aligned float2 -> single coalesced global b64 per fragment.
// ---------------------------------------------------------------------------
__global__ __launch_bounds__(256) void pack_weights_kernel(const float* __restrict__ w,
                                                           float* __restrict__ bp) {
  int i = blockIdx.x * 256 + threadIdx.x;   // over KTOT*CO = 93312
  if (i >= KTOT * CO) return;
  int n   = i % CO;
  int k   = i / CO;
  int cin = k % CI;
  int tap = k / CI;
  bp[(k >> 1) * (2 * CO) + 2 * n + (k & 1)] = w[n * KTOT + cin * 81 + tap];
}

// ---------------------------------------------------------------------------
// Main kernel: one block per (b, lo, do, ho); 96 threads = 3 waves,
// wave w owns cout tile n0 = 16*w and the three wo tiles {0,16,32}.
// ---------------------------------------------------------------------------
__global__ __launch_bounds__(96) void conv4d_wmma_kernel(const float* __restrict__ x,
                                                         const float* __restrict__ bp,
                                                         const float* __restrict__ bias3d,
                                                         float* __restrict__ out) {
  __shared__ float lds[LDSF];

  const int tid  = threadIdx.x;
  const int lane = tid & 31;
  const int wave = tid >> 5;      // 0..2  -> cout tile
  const int half = lane >> 4;     // lane half: K pair select / C row offset
  const int m    = lane & 15;     // row (A) / col (B,C,D) within tile

  int id = blockIdx.x;
  const int ho = id % HOUT; id /= HOUT;
  const int dd = id % DOUT; id /= DOUT;
  const int lo = id % LOUT;
  const int b  = id / LOUT;

  const int n0 = wave * 16;
  const int n  = n0 + m;          // this lane's output-channel column

  // bias summed over lk, folded into accumulator init (all 8 C elements of a
  // lane share column n)
  const float tb = bias3d[n] + bias3d[CO + n] + bias3d[2 * CO + n];
  v8f acc0, acc1, acc2;
  #pragma unroll
  for (int i = 0; i < 8; ++i) { acc0[i] = tb; acc1[i] = tb; acc2[i] = tb; }

  const float* xbase = x + b * BS + lo * LS + dd * DS2 + ho * HS;

  for (int cc = 0; cc < 3; ++cc) {          // C_IN chunks of 8
    const int C0 = cc * 8;
    __syncthreads();                        // protect LDS vs previous compute

    // ---- stage: transpose 8 channels x 27 taps x 48 w into [t3][w][cl]
    // units: (cin-pair p, t3, w-chunk of 4) = 4*27*12 = 1296
    const float* xc = xbase + C0 * CS;
    for (int u = tid; u < 1296; u += 96) {
      int wc = u % 12;
      int r  = u / 12;
      int t3 = r % 27;
      int p  = r / 27;
      int cl = 2 * p;
      int lk = t3 / 9;
      int dk = (t3 % 9) / 3;
      int hk = t3 % 3;
      const float* srcA = xc + cl * CS + lk * LS + dk * DS2 + hk * HS + wc * 4;
      if (cc < 2) __builtin_prefetch(srcA + 8 * CS, 0, 1);  // next chunk slab
      const float4 fa = *(const float4*)srcA;
      const float4 fb = *(const float4*)(srcA + CS);
      const float* pa = (const float*)&fa;
      const float* pbv = (const float*)&fb;
      #pragma unroll
      for (int j = 0; j < 4; ++j) {
        v2f pr; pr.x = pa[j]; pr.y = pbv[j];
        *(v2f*)&lds[(t3 * WP + (wc * 4 + j)) * PADC + cl] = pr;
      }
    }
    __syncthreads();

    // ---- implicit GEMM over this chunk: K = 27 taps * 3 wk * 8 cin
    for (int t3 = 0; t3 < 27; ++t3) {
      const float* arow0 = &lds[(t3 * WP + m) * PADC + 2 * half];
      #pragma unroll
      for (int wk = 0; wk < 3; ++wk) {
        #pragma unroll
        for (int c4 = 0; c4 < 8; c4 += 4) {
          const int k = (t3 * 3 + wk) * CI + C0 + c4;    // even
          // B fragment: rows {k+2h, k+2h+1}, col n -> one aligned b64
          const v2f bf = *(const v2f*)(bp + ((k >> 1) + half) * (2 * CO) + 2 * n);
          // A fragments: cin pair (c4+2h, c4+2h+1) at (t3, w0+m+wk) -> b64
          const float* abase = arow0 + wk * PADC + c4;
          const v2f a0 = *(const v2f*)(abase);                 // wo tile 0
          const v2f a1 = *(const v2f*)(abase + 16 * PADC);     // wo tile 1
          const v2f a2 = *(const v2f*)(abase + 32 * PADC);     // wo tile 2
          acc0 = __builtin_amdgcn_wmma_f32_16x16x4_f32(false, a0, false, bf, (short)0, acc0, false, false);
          acc1 = __builtin_amdgcn_wmma_f32_16x16x4_f32(false, a1, false, bf, (short)0, acc1, false, false);
          acc2 = __builtin_amdgcn_wmma_f32_16x16x4_f32(false, a2, false, bf, (short)0, acc2, false, false);
        }
      }
    }
  }

  // ---- store: lane holds col n, rows m = v + 8*half (consecutive wo)
  float* obase = out + ((((b * CO + n) * LOUT + lo) * DOUT + dd) * HOUT + ho) * WOUT;
  #pragma unroll
  for (int j = 0; j < 3; ++j) {
    v8f acc = (j == 0) ? acc0 : ((j == 1) ? acc1 : acc2);
    const int wo0 = j * 16 + 8 * half;
    #pragma unroll
    for (int v = 0; v < 8; ++v) {
      const int wo = wo0 + v;
      if (wo < WOUT) obase[wo] = acc[v];    // masks the partial third tile
    }
  }
}

// ---------------------------------------------------------------------------
extern "C" void kernel_launch(void* const* d_in, const int* in_sizes, int n_in,
                              void* d_out, int out_size, void* d_ws, size_t ws_size,
                              hipStream_t stream) {
  (void)in_sizes; (void)n_in; (void)out_size; (void)ws_size;
  const float* x      = (const float*)d_in[0];   // (2,24,16,16,48,48) f32
  const float* weight = (const float*)d_in[1];   // (48,24,3,3,3,3)   f32
  const float* bias3d = (const float*)d_in[2];   // (3,48)            f32
  float* out = (float*)d_out;                    // (2,48,14,14,46,46) f32
  float* bp  = (float*)d_ws;                     // needs 1944*48*4 = 373,248 B

  // 1) pack weights pair-interleaved for single-b64 B fragments
  const int npack = KTOT * CO;
  pack_weights_kernel<<<(npack + 255) / 256, 256, 0, stream>>>(weight, bp);

  // 2) implicit-GEMM conv: one block per (b, lo, do, ho)
  const int nblk = NBATCH * LOUT * DOUT * HOUT;  // 18032
  conv4d_wmma_kernel<<<nblk, 96, 0, stream>>>(x, bp, bias3d, out);
}